// GeometricTransformationLayer_65515431133592
// MI455X (gfx1250) — compile-verified
//
#include <hip/hip_runtime.h>

// Problem constants (B, W, H, D = 8, 128, 128, 128)
#define WDIM   128
#define NPT    (128 * 128 * 128)   // 2,097,152 points
#define NBATCH 8

typedef __attribute__((ext_vector_type(2))) float v2f;
typedef __attribute__((ext_vector_type(8))) float v8f;

// ---------------------------------------------------------------------------
// Prelude: M1 = T @ S @ R  -> ws[0..15];  M2 = T @ S -> ws[16..31]
// ---------------------------------------------------------------------------
__global__ void gtl_build_mats(const float* __restrict__ R,
                               const float* __restrict__ S,
                               const float* __restrict__ T,
                               float* __restrict__ ws) {
    if (threadIdx.x == 0 && blockIdx.x == 0) {
        float TS[16];
        for (int r = 0; r < 4; ++r)
            for (int c = 0; c < 4; ++c) {
                float a = 0.f;
                for (int k = 0; k < 4; ++k) a += T[r * 4 + k] * S[k * 4 + c];
                TS[r * 4 + c] = a;
            }
        for (int r = 0; r < 4; ++r)
            for (int c = 0; c < 4; ++c) {
                float a = 0.f;
                for (int k = 0; k < 4; ++k) a += TS[r * 4 + k] * R[k * 4 + c];
                ws[r * 4 + c] = a;                 // M1
            }
        for (int i = 0; i < 16; ++i) ws[16 + i] = TS[i];  // M2
    }
}

// ---------------------------------------------------------------------------
// Main kernel: per wave, 2 chained V_WMMA_F32_16X16X4_F32 compute the warped
// coordinates of 16 points (p = M1*q + M2*ct); then all 32 lanes gather:
// lane half selects batches 0-3 vs 4-7, lane&15 selects the point.
// ---------------------------------------------------------------------------
__global__ __launch_bounds__(256)
void gtl_warp_gather(const float* __restrict__ x,     // [8, NPT]
                     const float* __restrict__ pts,   // [4, NPT]
                     const float* __restrict__ ct,    // [4, NPT]
                     const float* __restrict__ mats,  // M1(16) + M2(16)
                     float* __restrict__ out) {       // [8, NPT]
    const int lane = threadIdx.x & 31;
    const int wave = threadIdx.x >> 5;
    const int n0   = (blockIdx.x * 8 + wave) * 16;    // 16 points per wave
    const int ncol = lane & 15;                       // N column / A row / point
    const int hi   = lane >> 4;                       // lane half
    const int kb   = hi * 2;                          // K base for this half

    // --- A operands: 16x4, rows 0..3 = M rows, rows 4..15 zero -------------
    const int mr = (ncol < 4) ? ncol : 0;
    float a1x = mats[mr * 4 + kb],      a1y = mats[mr * 4 + kb + 1];
    float a2x = mats[16 + mr * 4 + kb], a2y = mats[16 + mr * 4 + kb + 1];
    v2f a1, a2;
    a1.x = (ncol < 4) ? a1x : 0.f;  a1.y = (ncol < 4) ? a1y : 0.f;
    a2.x = (ncol < 4) ? a2x : 0.f;  a2.y = (ncol < 4) ? a2y : 0.f;

    // --- B operands: 4x16 columns = homogeneous point / ct vectors ---------
    v2f bp, bc;
    bp.x = pts[(size_t)(kb + 0) * NPT + n0 + ncol];
    bp.y = pts[(size_t)(kb + 1) * NPT + n0 + ncol];
    bc.x = ct [(size_t)(kb + 0) * NPT + n0 + ncol];
    bc.y = ct [(size_t)(kb + 1) * NPT + n0 + ncol];

    // --- p = M1*q + M2*ct via chained WMMA ---------------------------------
    v8f cz = {};
    v8f d;
    d = __builtin_amdgcn_wmma_f32_16x16x4_f32(false, a1, false, bp,
                                              (short)0, cz, false, false);
    d = __builtin_amdgcn_wmma_f32_16x16x4_f32(false, a2, false, bc,
                                              (short)0, d,  false, false);

    // D rows 0..2 (pi,pj,pk) live in lanes 0..15; broadcast to both halves.
    const float pi = __shfl(d[0], ncol, 32);
    const float pj = __shfl(d[1], ncol, 32);
    const float pk = __shfl(d[2], ncol, 32);

    // --- trilinear corner weights (reference clamp semantics) --------------
    const float WM = (float)(WDIM - 1);
    const float ci_ = fminf(fmaxf(pi, 0.f), WM);
    const float cj_ = fminf(fmaxf(pj, 0.f), WM);
    const float ck_ = fminf(fmaxf(pk, 0.f), WM);
    const float fi = floorf(pi), fj = floorf(pj), fk = floorf(pk);

    const float i0 = fminf(fmaxf(fi,       0.f), WM);
    const float i1 = fminf(fmaxf(fi + 1.f, 0.f), WM);
    const float j0 = fminf(fmaxf(fj,       0.f), WM);
    const float j1 = fminf(fmaxf(fj + 1.f, 0.f), WM);
    const float k0 = fminf(fmaxf(fk,       0.f), WM);
    const float k1 = fminf(fmaxf(fk + 1.f, 0.f), WM);

    const float wi0 = 1.f - fabsf(ci_ - i0), wi1 = 1.f - fabsf(ci_ - i1);
    const float wj0 = 1.f - fabsf(cj_ - j0), wj1 = 1.f - fabsf(cj_ - j1);
    const float wk0 = 1.f - fabsf(ck_ - k0), wk1 = 1.f - fabsf(ck_ - k1);

    const int ii0 = (int)i0 * (WDIM * WDIM), ii1 = (int)i1 * (WDIM * WDIM);
    const int jj0 = (int)j0 * WDIM,          jj1 = (int)j1 * WDIM;
    const int kk0 = (int)k0,                 kk1 = (int)k1;

    const float wtot = (wi0 + wi1) * (wj0 + wj1) * (wk0 + wk1);
    const float inv  = 1.f / fminf(fmaxf(wtot, 1e-8f), 8.0f);

    int   idx[8];
    float wgt[8];
    idx[0] = ii0 + jj0 + kk0; wgt[0] = wi0 * wj0 * wk0;
    idx[1] = ii1 + jj0 + kk0; wgt[1] = wi1 * wj0 * wk0;
    idx[2] = ii0 + jj1 + kk0; wgt[2] = wi0 * wj1 * wk0;
    idx[3] = ii1 + jj1 + kk0; wgt[3] = wi1 * wj1 * wk0;
    idx[4] = ii0 + jj0 + kk1; wgt[4] = wi0 * wj0 * wk1;
    idx[5] = ii1 + jj0 + kk1; wgt[5] = wi1 * wj0 * wk1;
    idx[6] = ii0 + jj1 + kk1; wgt[6] = wi0 * wj1 * wk1;
    idx[7] = ii1 + jj1 + kk1; wgt[7] = wi1 * wj1 * wk1;

    // --- gather: this lane handles 4 batches of point n --------------------
    const int    n     = n0 + ncol;
    const int    bbase = hi * 4;                       // 0..3 or 4..7
    const float* xb    = x + (size_t)bbase * NPT;
    float acc0 = 0.f, acc1 = 0.f, acc2 = 0.f, acc3 = 0.f;
#pragma unroll
    for (int c = 0; c < 8; ++c) {
        const float w  = wgt[c];
        const int   id = idx[c];
        acc0 = fmaf(w, xb[id],                    acc0);
        acc1 = fmaf(w, xb[(size_t)1 * NPT + id],  acc1);
        acc2 = fmaf(w, xb[(size_t)2 * NPT + id],  acc2);
        acc3 = fmaf(w, xb[(size_t)3 * NPT + id],  acc3);
    }

    float* ob = out + (size_t)bbase * NPT + n;
    ob[0]                 = acc0 * inv;
    ob[(size_t)1 * NPT]   = acc1 * inv;
    ob[(size_t)2 * NPT]   = acc2 * inv;
    ob[(size_t)3 * NPT]   = acc3 * inv;
}

// ---------------------------------------------------------------------------
extern "C" void kernel_launch(void* const* d_in, const int* in_sizes, int n_in,
                              void* d_out, int out_size, void* d_ws, size_t ws_size,
                              hipStream_t stream) {
    (void)in_sizes; (void)n_in; (void)out_size; (void)ws_size;
    const float* x   = (const float*)d_in[0];
    const float* pts = (const float*)d_in[1];
    const float* ct  = (const float*)d_in[2];
    const float* R   = (const float*)d_in[3];
    const float* S   = (const float*)d_in[4];
    const float* T   = (const float*)d_in[5];
    float*       out = (float*)d_out;
    float*       ws  = (float*)d_ws;   // 32 floats used

    gtl_build_mats<<<1, 32, 0, stream>>>(R, S, T, ws);

    const int blocks = NPT / 128;      // 16384 blocks, 8 waves x 16 pts each
    gtl_warp_gather<<<blocks, 256, 0, stream>>>(x, pts, ct, ws, out);
}